// GRUNet_2594160247035
// MI455X (gfx1250) — compile-verified
//
#include <hip/hip_runtime.h>
#include <hip/hip_bf16.h>
#include <math.h>

// -------------------------------------------------------------------------
// GRU step + vocab projection for MI455X (gfx1250).
// Memory-bound: ~512MB fp32 weight streaming per call -> ~22us floor @23.3TB/s.
// fp32 WMMA (V_WMMA_F32_16X16X4_F32) for the GEMVs (precision-preserving; WMMA
// rate is never the bottleneck at AI ~ 0.5 FLOP/byte), with CDNA5 async
// global->LDS (GLOBAL_LOAD_ASYNC_TO_LDS_B128, ASYNCcnt) double-buffering the
// weight tiles so the HBM stream overlaps the WMMA k-loop.
// -------------------------------------------------------------------------

#define H       2048
#define H3      6144
#define VOCAB   50257
#define TILE_K  64
#define ROWS_PB 64            // rows per block = 4 waves * 16
#define LDS_STR 68            // 68 floats = 272B rows: 16B-aligned b128 dsts,
                              // and bank = (4r + c) % 64 -> conflict-free frags
#define TSZ     (ROWS_PB * LDS_STR)
#define NT      (H / TILE_K)  // 32 k-tiles

typedef float v2f __attribute__((ext_vector_type(2)));
typedef float v8f __attribute__((ext_vector_type(8)));
typedef int   v4i __attribute__((ext_vector_type(4)));

#define GLOBAL_AS __attribute__((address_space(1)))
#define LDS_AS    __attribute__((address_space(3)))

typedef GLOBAL_AS v4i* gv4i_ptr;   // int4 __device__ *  (per clang diagnostic)
typedef LDS_AS    v4i* lv4i_ptr;   // int4 __shared__ *

#if defined(__AMDGCN__) && __has_builtin(__builtin_amdgcn_global_load_async_to_lds_b128)
#define HAVE_ASYNC 1
#else
#define HAVE_ASYNC 0
#endif

#if HAVE_ASYNC
# if __has_builtin(__builtin_amdgcn_s_wait_asynccnt)
#  define WAIT_ASYNC(n) __builtin_amdgcn_s_wait_asynccnt(n)
# else
#  define WAIT_ASYNC(n) asm volatile("s_wait_asynccnt %0" :: "n"(n) : "memory")
# endif
#else
# define WAIT_ASYNC(n) ((void)0)
#endif

// ---- issue one 64x64 K-tile of W (+ the matching v chunk) into LDS ------
// 1024 float4 transfers; 8 per thread. Async path: 8 ASYNCcnt ops per wave.
__device__ __forceinline__ void issue_tile(
    const float* __restrict__ W, const float* __restrict__ v,
    int N, int n_block, int k0, float* tile, float* vsh, int t)
{
    if (t < TILE_K) vsh[t] = v[k0 + t];
    #pragma unroll
    for (int i = 0; i < 8; ++i) {
        int f  = t + i * 128;
        int r  = f >> 4;           // tile row 0..63
        int c4 = f & 15;           // float4 column 0..15
        int gr = n_block + r;
        if (gr >= N) gr = N - 1;   // clamp (tail rows never written out)
        const float* gsrc = W + (size_t)gr * H + k0 + (c4 << 2);
        float*       ldst = &tile[r * LDS_STR + (c4 << 2)];
#if HAVE_ASYNC
        __builtin_amdgcn_global_load_async_to_lds_b128(
            (gv4i_ptr)gsrc, (lv4i_ptr)ldst, 0, 0);
#else
        const float4 w4 = *(const float4*)gsrc;
        ldst[0] = w4.x; ldst[1] = w4.y; ldst[2] = w4.z; ldst[3] = w4.w;
        if (k0 + TILE_K < H)       // stream next K-tile: global_prefetch_b8
            __builtin_prefetch(gsrc + TILE_K, 0, 1);
#endif
    }
}

// ---- WMMA GEMV core: y[n] = dot(v, W[n,:]) + bias[n], K = H = 2048 -------
// Block: 128 threads (4 waves); each wave produces 16 output rows.
// Fragment layouts for V_WMMA_F32_16X16X4_F32 (wave32):
//   A (16x4): VGPR0 = {K0 lanes0-15 | K2 lanes16-31}, VGPR1 = {K1 | K3} (ISA)
//   B (4x16): mirrored -> VGPR0 = {K0 | K2}, VGPR1 = {K1 | K3}, N = lane&15
//   C row M=0 lives in c[0], lanes 0-15, N = lane.
__device__ __forceinline__ void gemv_wmma_body(
    const float* __restrict__ W, const float* __restrict__ v,
    const float* __restrict__ bias, float* __restrict__ y,
    int N, int n_block, float* tile2, float* vsh2)
{
    const int t     = threadIdx.x;     // 0..127
    const int lane  = t & 31;
    const int wave  = t >> 5;          // 0..3
    const int nlane = lane & 15;       // N index / M index
    const int half  = lane >> 4;       // 0 or 1

    v8f acc = {};

    // prologue: stage tile 0
    issue_tile(W, v, N, n_block, 0, tile2, vsh2, t);

    for (int it = 0; it < NT; ++it) {
        const int cur = it & 1;
        float* tile = tile2 + cur * TSZ;
        float* vsh  = vsh2  + cur * TILE_K;

        if (it + 1 < NT) {
            issue_tile(W, v, N, n_block, (it + 1) * TILE_K,
                       tile2 + (cur ^ 1) * TSZ, vsh2 + (cur ^ 1) * TILE_K, t);
            WAIT_ASYNC(8);             // 8 outstanding = next tile only
        } else {
            WAIT_ASYNC(0);
        }
        __syncthreads();               // current tile visible WGP-wide

        #pragma unroll
        for (int kk = 0; kk < TILE_K; kk += 4) {
            v2f b;                     // B from LDS (conflict-free banking)
            b.x = tile[nlane * LDS_STR + kk + 2 * half];       // K0 | K2
            b.y = tile[nlane * LDS_STR + kk + 1 + 2 * half];   // K1 | K3
            v2f a;                     // only M=0 carries data; branch-free
            a.x = (nlane == 0) ? vsh[kk + 2 * half]     : 0.0f;
            a.y = (nlane == 0) ? vsh[kk + 1 + 2 * half] : 0.0f;
            acc = __builtin_amdgcn_wmma_f32_16x16x4_f32(
                    false, a, false, b, (short)0, acc, false, false);
        }
        __syncthreads();               // protect buffer reuse at it+1's issue
    }

    if (lane < 16) {                   // C: VGPR0, lanes 0-15 = (M=0, N=lane)
        int n = n_block + wave * 16 + lane;
        if (n < N) y[n] = acc[0] + bias[n];
    }
}

__global__ __launch_bounds__(128) void gemv_wmma(
    const float* __restrict__ W, const float* __restrict__ v,
    const float* __restrict__ bias, float* __restrict__ y, int N)
{
    __shared__ float tile2[2 * TSZ];   // ~34.8 KB (of 320KB/WGP)
    __shared__ float vsh2[2 * TILE_K];
    gemv_wmma_body(W, v, bias, y, N, blockIdx.x * ROWS_PB, tile2, vsh2);
}

// Both gate GEMVs in one launch for occupancy (blocks 0..95: Wx*x, 96..191: Wh*h)
__global__ __launch_bounds__(128) void gemv_wmma_dual(
    const float* __restrict__ Wx, const float* __restrict__ xv,
    const float* __restrict__ bx, float* __restrict__ xs,
    const float* __restrict__ Wh, const float* __restrict__ hv,
    const float* __restrict__ bh, float* __restrict__ hs, int N)
{
    __shared__ float tile2[2 * TSZ];
    __shared__ float vsh2[2 * TILE_K];
    const int nb = (N + ROWS_PB - 1) / ROWS_PB;      // 96
    const bool second = (int)blockIdx.x >= nb;
    const int n_block = (blockIdx.x - (second ? nb : 0)) * ROWS_PB;
    gemv_wmma_body(second ? Wh : Wx, second ? hv : xv,
                   second ? bh : bx, second ? hs : xs,
                   N, n_block, tile2, vsh2);
}

// ---- x = relu(embedding[c]) ---------------------------------------------
__global__ void embed_relu(const int* __restrict__ ch,
                           const float* __restrict__ emb,
                           float* __restrict__ x)
{
    int i = blockIdx.x * blockDim.x + threadIdx.x;
    if (i < H) {
        int c = ch[0];
        x[i] = fmaxf(emb[(size_t)c * H + i], 0.0f);
    }
}

// ---- GRU gates: h = (1-z)*n + z*h_prev ----------------------------------
__global__ void gru_gates(const float* __restrict__ xs,
                          const float* __restrict__ hs,
                          const float* __restrict__ hprev,
                          float* __restrict__ hout)
{
    int i = blockIdx.x * blockDim.x + threadIdx.x;
    if (i < H) {
        float r = 1.0f / (1.0f + expf(-(xs[i]       + hs[i])));
        float z = 1.0f / (1.0f + expf(-(xs[H + i]   + hs[H + i])));
        float n = tanhf(xs[2 * H + i] + r * hs[2 * H + i]);
        hout[i] = (1.0f - z) * n + z * hprev[i];
    }
}

// ---- log-softmax, stage 1: per-block (max, sum exp) ---------------------
__global__ __launch_bounds__(256) void softmax_partial(
    const float* __restrict__ logits, float* __restrict__ red, int N)
{
    __shared__ float sm[256];
    int t = threadIdx.x;
    float m = -3.4e38f;
    for (int i = blockIdx.x * 256 + t; i < N; i += gridDim.x * 256)
        m = fmaxf(m, logits[i]);
    sm[t] = m; __syncthreads();
    for (int s = 128; s > 0; s >>= 1) {
        if (t < s) sm[t] = fmaxf(sm[t], sm[t + s]);
        __syncthreads();
    }
    float bm = sm[0]; __syncthreads();
    float sum = 0.0f;
    for (int i = blockIdx.x * 256 + t; i < N; i += gridDim.x * 256)
        sum += expf(logits[i] - bm);
    sm[t] = sum; __syncthreads();
    for (int s = 128; s > 0; s >>= 1) {
        if (t < s) sm[t] += sm[t + s];
        __syncthreads();
    }
    if (t == 0) { red[2 * blockIdx.x] = bm; red[2 * blockIdx.x + 1] = sm[0]; }
}

// ---- log-softmax, stage 2: combine 64 partials + write output -----------
__global__ void softmax_final(const float* __restrict__ logits,
                              const float* __restrict__ red,
                              float* __restrict__ out, int N, int nparts)
{
    float M = -3.4e38f;
    for (int p = 0; p < nparts; ++p) M = fmaxf(M, red[2 * p]);
    float S = 0.0f;
    for (int p = 0; p < nparts; ++p) S += red[2 * p + 1] * expf(red[2 * p] - M);
    float ls = M + logf(S);
    int i = blockIdx.x * blockDim.x + threadIdx.x;
    if (i < N) out[i] = logits[i] - ls;
}

// -------------------------------------------------------------------------
extern "C" void kernel_launch(void* const* d_in, const int* in_sizes, int n_in,
                              void* d_out, int out_size, void* d_ws, size_t ws_size,
                              hipStream_t stream)
{
    const int*   input_char = (const int*)  d_in[0];
    const float* hidden     = (const float*)d_in[1];   // (1,1,2048)
    const float* embedding  = (const float*)d_in[2];   // (50257,2048)
    const float* weightx    = (const float*)d_in[3];   // (6144,2048)
    const float* weighth    = (const float*)d_in[4];   // (6144,2048)
    const float* biasx      = (const float*)d_in[5];   // (6144,)
    const float* biash      = (const float*)d_in[6];   // (6144,)
    const float* W_out      = (const float*)d_in[7];   // (50257,2048)
    const float* b_out      = (const float*)d_in[8];   // (50257,)

    float* out  = (float*)d_out;                       // [0..50256] log_softmax
    float* hout = out + VOCAB;                         // [50257..52304] new h

    // workspace (floats): x(2048) | xs(6144) | hs(6144) | logits(50304) | red(128)
    float* ws     = (float*)d_ws;
    float* xvec   = ws;
    float* xs     = ws + H;
    float* hs     = ws + H + H3;
    float* logits = ws + H + 2 * H3;
    float* red    = logits + 50304;                    // ~259 KB total

    // 1) x = relu(embedding[c])
    embed_relu<<<(H + 255) / 256, 256, 0, stream>>>(input_char, embedding, xvec);

    // 2) xs = Wx x + bx ; hs = Wh h + bh   (two 6144x2048 fp32 WMMA GEMVs)
    const int nbGate = (H3 + ROWS_PB - 1) / ROWS_PB;   // 96
    gemv_wmma_dual<<<2 * nbGate, 128, 0, stream>>>(
        weightx, xvec, biasx, xs, weighth, hidden, biash, hs, H3);

    // 3) gates -> h (written straight into d_out tail)
    gru_gates<<<(H + 255) / 256, 256, 0, stream>>>(xs, hs, hidden, hout);

    // 4) logits = W_out h + b_out   (50257x2048 fp32 WMMA GEMV, 412MB stream)
    const int nbOut = (VOCAB + ROWS_PB - 1) / ROWS_PB; // 786
    gemv_wmma<<<nbOut, 128, 0, stream>>>(W_out, hout, b_out, logits, VOCAB);

    // 5) log_softmax
    softmax_partial<<<64, 256, 0, stream>>>(logits, red, VOCAB);
    softmax_final<<<(VOCAB + 255) / 256, 256, 0, stream>>>(logits, red, out, VOCAB, 64);
}